// LightGCN_layer_79774722556256
// MI455X (gfx1250) — compile-verified
//
#include <hip/hip_runtime.h>
#include <cstdint>

// LightGCN propagation: out[row[e]] += all_emb[col[e]] * w[e]
// Memory-bound scatter/gather: ~2 GB gather (8x row reuse, partially L2-
// resident) + ~2 GB atomic-add traffic + 0.25 GB zero-fill -> O(150us) at
// 23.3 TB/s. CDNA5 path: async gather engine (global_load_async_to_lds_b64,
// ASYNCcnt pipelining) + hardware no-return global_atomic_add_f32 scatter.
//
// NOTE: scatter kernel is defined FIRST so the disasm snippet shows it.

#define WAVE 32
#define WPB  8                 // waves per block
#define BLOCK (WAVE * WPB)
#define K    8                 // async pipeline depth (edges in flight / wave)
#define EPC  32                // edges per chunk (one chunk per wave)

// Async copy of 8 bytes/lane from global (64-bit VGPR address, GV mode) into
// LDS. Default RT temporal hint on purpose: embedding rows see ~8x reuse
// across edges, so we want them cached in L2, not streamed.
static __device__ __forceinline__ void async_gather_b64(const void* gaddr,
                                                        unsigned lds_addr) {
    asm volatile("global_load_async_to_lds_b64 %0, %1, off"
                 :: "v"(lds_addr), "v"(gaddr)
                 : "memory");
}

// Wait until <= n async ops outstanding (SIMM16 must be a literal; the fast
// path is fully unrolled so this folds to one s_wait_asynccnt).
static __device__ __forceinline__ void wait_async_le(int n) {
    switch (n) {
    case 0:  asm volatile("s_wait_asynccnt 0" ::: "memory"); break;
    case 1:  asm volatile("s_wait_asynccnt 1" ::: "memory"); break;
    case 2:  asm volatile("s_wait_asynccnt 2" ::: "memory"); break;
    case 3:  asm volatile("s_wait_asynccnt 3" ::: "memory"); break;
    case 4:  asm volatile("s_wait_asynccnt 4" ::: "memory"); break;
    case 5:  asm volatile("s_wait_asynccnt 5" ::: "memory"); break;
    case 6:  asm volatile("s_wait_asynccnt 6" ::: "memory"); break;
    default: asm volatile("s_wait_asynccnt 7" ::: "memory"); break;
    }
}

static __device__ __forceinline__ const float*
row_base(int c, const float* users, const float* items, int num_users) {
    return (c < num_users) ? (users + (long long)c * 64)
                           : (items + (long long)(c - num_users) * 64);
}

__global__ void __launch_bounds__(BLOCK)
lightgcn_scatter_kernel(const float* __restrict__ users,
                        const float* __restrict__ items,
                        const int*  __restrict__ rows,
                        const int*  __restrict__ cols,
                        const float* __restrict__ vals,
                        float* __restrict__ out,
                        int num_edges, int num_users) {
    __shared__ float2 lbuf[WPB][K][WAVE];   // 16 KB / block

    const int lane = threadIdx.x & (WAVE - 1);
    const int warp = threadIdx.x >> 5;
    const long long e0 = ((long long)blockIdx.x * WPB + warp) * EPC;
    if (e0 >= num_edges) return;
    const long long rem = (long long)num_edges - e0;
    const int count = rem < EPC ? (int)rem : EPC;

    // Lane l caches metadata for edge e0+l; broadcast later via shfl.
    int myRow = 0, myCol = 0;
    float myVal = 0.f;
    if (lane < count) {
        myRow = rows[e0 + lane];
        myCol = cols[e0 + lane];
        myVal = vals[e0 + lane];
    }

    if (count == EPC) {
        // ---- Fast path: full 32-edge chunk, K-deep async pipeline. ----
        const unsigned lbase = (unsigned)(uintptr_t)&lbuf[warp][0][lane];
        // stage stride = WAVE * sizeof(float2) = 256 bytes

#pragma unroll
        for (int s = 0; s < K; ++s) {
            int c = __shfl(myCol, s, WAVE);
            async_gather_b64(row_base(c, users, items, num_users) + lane * 2,
                             lbase + (unsigned)s * 256u);
        }

#pragma unroll
        for (int j = 0; j < EPC; ++j) {
            // issued = min(EPC, j+K); edge j complete <=> outstanding <=
            // issued-(j+1) == min(K-1, EPC-1-j). Constant after unroll.
            wait_async_le((j <= EPC - K) ? (K - 1) : (EPC - 1 - j));

            float2 m = lbuf[warp][j & (K - 1)][lane];
            float v  = __shfl(myVal, j, WAVE);
            int   r  = __shfl(myRow, j, WAVE);

            float* dst = out + (long long)r * 64 + lane * 2;
            // Relaxed, agent scope -> no-return global_atomic_add_f32.
            (void)__hip_atomic_fetch_add(dst,     m.x * v, __ATOMIC_RELAXED,
                                         __HIP_MEMORY_SCOPE_AGENT);
            (void)__hip_atomic_fetch_add(dst + 1, m.y * v, __ATOMIC_RELAXED,
                                         __HIP_MEMORY_SCOPE_AGENT);

            // Refill the stage we just consumed. The loaded value fed the
            // atomics above (s_wait_dscnt satisfied before the async issue in
            // program order), so this LDS WAR is safe.
            if (j + K < EPC) {
                int c = __shfl(myCol, j + K, WAVE);
                async_gather_b64(
                    row_base(c, users, items, num_users) + lane * 2,
                    lbase + (unsigned)(j & (K - 1)) * 256u);
            }
        }
    } else {
        // ---- Tail path (never taken for E % 32 == 0): direct gather. ----
        for (int j = 0; j < count; ++j) {
            int   c = __shfl(myCol, j, WAVE);
            int   r = __shfl(myRow, j, WAVE);
            float v = __shfl(myVal, j, WAVE);
            const float2* src = (const float2*)(row_base(c, users, items,
                                                         num_users)) + lane;
            float2 m = *src;
            float* dst = out + (long long)r * 64 + lane * 2;
            (void)__hip_atomic_fetch_add(dst,     m.x * v, __ATOMIC_RELAXED,
                                         __HIP_MEMORY_SCOPE_AGENT);
            (void)__hip_atomic_fetch_add(dst + 1, m.y * v, __ATOMIC_RELAXED,
                                         __HIP_MEMORY_SCOPE_AGENT);
        }
    }
}

__global__ void zero_f4_kernel(float4* __restrict__ out, long long n4) {
    long long i = (long long)blockIdx.x * blockDim.x + threadIdx.x;
    if (i < n4) out[i] = make_float4(0.f, 0.f, 0.f, 0.f);
}

extern "C" void kernel_launch(void* const* d_in, const int* in_sizes, int n_in,
                              void* d_out, int out_size, void* d_ws, size_t ws_size,
                              hipStream_t stream) {
    const float* users = (const float*)d_in[0];
    const float* items = (const float*)d_in[1];
    const int*   eidx  = (const int*)d_in[2];   // edge_index, shape (2, E) flat
    const float* evals = (const float*)d_in[3];
    float* out = (float*)d_out;

    const int num_users = in_sizes[0] / 64;
    const int num_edges = in_sizes[3];
    const int* rows = eidx;
    const int* cols = eidx + num_edges;

    // 1) Zero the accumulator every launch (graph replays accumulate atomically).
    const long long n4 = (long long)out_size / 4;
    const int zb = (int)((n4 + 255) / 256);
    zero_f4_kernel<<<zb, 256, 0, stream>>>((float4*)out, n4);

    // 2) Async-pipelined gather + atomic scatter-add. One wave per 32 edges.
    const long long chunks = ((long long)num_edges + EPC - 1) / EPC;
    const int sb = (int)((chunks + WPB - 1) / WPB);
    lightgcn_scatter_kernel<<<sb, BLOCK, 0, stream>>>(
        users, items, rows, cols, evals, out, num_edges, num_users);
}